// GraphConstructor_69793218560438
// MI455X (gfx1250) — compile-verified
//
#include <hip/hip_runtime.h>
#include <hip/hip_bf16.h>

typedef __attribute__((ext_vector_type(2))) float v2f;
typedef __attribute__((ext_vector_type(8))) float v8f;

#define NNODES 390
#define BSZ 32
#define EDGES_PER_B (NNODES * NNODES)          // 152100
#define NODE_OUT_ELEMS (BSZ * NNODES * 64)     // 798720
#define EDGE_OUT_ELEMS (BSZ * EDGES_PER_B * 32)// 155750400

// ---------------------------------------------------------------------------
// Kernel 1: fold the edge-path parameter chain into alpha[32], beta[32].
//   alpha[o] = sum_i edge_W[o,i] * pw_W[i]
//   beta[o]  = sum_i edge_W[o,i] * pw_b[i] + edge_b[o]
// ---------------------------------------------------------------------------
__global__ __launch_bounds__(32) void fold_kernel(
    const float* __restrict__ pw_W, const float* __restrict__ pw_b,
    const float* __restrict__ edge_W, const float* __restrict__ edge_b,
    float* __restrict__ ab /* [64] */) {
  int o = threadIdx.x;  // 0..31
  float a = 0.f, b = 0.f;
#pragma unroll
  for (int i = 0; i < 32; ++i) {
    float w = edge_W[o * 32 + i];
    a = fmaf(w, pw_W[i], a);
    b = fmaf(w, pw_b[i], b);
  }
  ab[o] = a;
  ab[32 + o] = b + edge_b[o];
}

// ---------------------------------------------------------------------------
// Kernel 2: node path via f32 WMMA.
//   feat1[m,k] = bias(m)*pb_W[k] + pb_b[k]           (m = b*390+node, K=64)
//   out[m,n]   = sum_k feat1[m,k]*node_W[n,k] + node_b[n] + pe[row(node),n]
// One wave per 16-row M tile; 4 N-subtiles of 16; 16 K-steps of 4 via
// V_WMMA_F32_16X16X4_F32 (D = A(16x4) x B(4x16) + C).
// ---------------------------------------------------------------------------
__global__ __launch_bounds__(32) void node_wmma_kernel(
    const float* __restrict__ b0, const float* __restrict__ b1,
    const float* __restrict__ b2, const float* __restrict__ b3,
    const float* __restrict__ pb_W, const float* __restrict__ pb_b,
    const float* __restrict__ node_W, const float* __restrict__ node_b,
    const float* __restrict__ pos_embed, float* __restrict__ out) {
  const int lane = threadIdx.x;
  const int half = lane >> 4;       // 0: K pair {0,1}; 1: K pair {2,3}
  const int l16  = lane & 15;

  // --- bias value for this lane's A-row (constant over K) ---
  const int mA   = blockIdx.x * 16 + l16;
  const int bb   = mA / NNODES;
  const int node = mA - bb * NNODES;
  float bv = 0.f;
  if (node >= 3) {
    int idx = node - 3;
    if (idx < 128)      bv = b0[bb * 128 + idx];
    else if (idx < 256) bv = b1[bb * 128 + (idx - 128)];
    else if (idx < 384) bv = b2[bb * 128 + (idx - 256)];
    else                bv = b3[bb * 3 + (idx - 384)];
  }

  v8f acc[4];
#pragma unroll
  for (int t = 0; t < 4; ++t) acc[t] = (v8f){};

#pragma unroll
  for (int k0 = 0; k0 < 64; k0 += 4) {
    const int ka = k0 + half * 2;
    // A fragment: lanes 0-15 hold M=lane, K=k0..k0+1; lanes 16-31 M=lane-16, K=k0+2..k0+3
    v2f a;
    a.x = fmaf(bv, pb_W[ka],     pb_b[ka]);
    a.y = fmaf(bv, pb_W[ka + 1], pb_b[ka + 1]);
#pragma unroll
    for (int t = 0; t < 4; ++t) {
      // B fragment: B[k][n] = node_W[n*64 + k]; lanes 0-15 hold K=k0..k0+1 (N=l16),
      // lanes 16-31 hold K=k0+2..k0+3.
      const int nB = t * 16 + l16;
      v2f bf;
      bf.x = node_W[nB * 64 + ka];
      bf.y = node_W[nB * 64 + ka + 1];
      acc[t] = __builtin_amdgcn_wmma_f32_16x16x4_f32(
          /*neg_a=*/false, a, /*neg_b=*/false, bf,
          /*c_mod=*/(short)0, acc[t], /*reuse_a=*/false, /*reuse_b=*/false);
    }
  }

  // --- epilogue: + node_b + pos_embed, store ---
  const int mBase = blockIdx.x * 16 + half * 8;  // C/D: VGPR r -> M = r (+8 for hi lanes)
#pragma unroll
  for (int t = 0; t < 4; ++t) {
    const int nD = t * 16 + l16;
    const float nb = node_b[nD];
#pragma unroll
    for (int r = 0; r < 8; ++r) {
      const int mD  = mBase + r;
      const int bD  = mD / NNODES;
      const int ndD = mD - bD * NNODES;
      const int prow = (ndD < 3) ? ndD
                     : (ndD < 131) ? 3
                     : (ndD < 259) ? 4
                     : (ndD < 387) ? 5
                     : (ndD - 381);
      out[(size_t)mD * 64 + nD] = acc[t][r] + nb + pos_embed[prow * 64 + nD];
    }
  }
}

// ---------------------------------------------------------------------------
// Kernel 3: edge path, fully folded:  out[e,o] = e*alpha[o] + beta[o].
// 8 lanes per edge; lane j handles channel quad j -> contiguous 512B per
// wave per store instruction. Lane j==0 also writes the mask element.
// ---------------------------------------------------------------------------
__global__ __launch_bounds__(256) void edge_kernel(
    const float* __restrict__ w0, const float* __restrict__ w1,
    const float* __restrict__ w2, const float* __restrict__ w3,
    const float* __restrict__ ab,
    float* __restrict__ edge_out, float* __restrict__ mask_out) {
  const unsigned tid  = blockIdx.x * 256u + threadIdx.x;
  const unsigned edge = tid >> 3;
  const unsigned j    = tid & 7u;

  const unsigned b   = edge / (unsigned)EDGES_PER_B;
  const unsigned rem = edge - b * (unsigned)EDGES_PER_B;
  const unsigned r   = rem / (unsigned)NNODES;
  const unsigned c   = rem - r * (unsigned)NNODES;

  // Band classification per _batch_to_graphs (reshape, not transpose!)
  float e = 0.f;
  if (r < 3u) {
    if (c >= 3u && c < 131u)        e = w0[b * 384u   + r * 128u         + (c - 3u)];
  } else if (r < 131u) {
    if (c >= 131u && c < 259u)      e = w1[b * 16384u + (r - 3u) * 128u  + (c - 131u)];
  } else if (r < 259u) {
    if (c >= 259u && c < 387u)      e = w2[b * 16384u + (r - 131u) * 128u + (c - 259u)];
  } else {
    if (c >= 387u)                  e = w3[b * 384u   + (r - 259u) * 3u  + (c - 387u)];
  }

  const float4 a4 = reinterpret_cast<const float4*>(ab)[j];
  const float4 b4 = reinterpret_cast<const float4*>(ab + 32)[j];
  float4 o;
  o.x = fmaf(e, a4.x, b4.x);
  o.y = fmaf(e, a4.y, b4.y);
  o.z = fmaf(e, a4.z, b4.z);
  o.w = fmaf(e, a4.w, b4.w);
  reinterpret_cast<float4*>(edge_out)[(size_t)edge * 8u + j] = o;

  if (j == 0u) mask_out[edge] = (e != 0.f) ? 1.0f : 0.0f;
}

// ---------------------------------------------------------------------------
extern "C" void kernel_launch(void* const* d_in, const int* in_sizes, int n_in,
                              void* d_out, int out_size, void* d_ws, size_t ws_size,
                              hipStream_t stream) {
  const float* w0 = (const float*)d_in[0];
  const float* w1 = (const float*)d_in[1];
  const float* w2 = (const float*)d_in[2];
  const float* w3 = (const float*)d_in[3];
  const float* b0 = (const float*)d_in[4];
  const float* b1 = (const float*)d_in[5];
  const float* b2 = (const float*)d_in[6];
  const float* b3 = (const float*)d_in[7];
  // d_in[8] = labels (unused)
  const float* pw_W   = (const float*)d_in[9];
  const float* pw_b   = (const float*)d_in[10];
  const float* pb_W   = (const float*)d_in[11];
  const float* pb_b   = (const float*)d_in[12];
  const float* node_W = (const float*)d_in[13];
  const float* node_b = (const float*)d_in[14];
  const float* edge_W = (const float*)d_in[15];
  const float* edge_b = (const float*)d_in[16];
  const float* pos_embed = (const float*)d_in[17];

  float* node_out = (float*)d_out;                       // (32,390,64)
  float* edge_out = node_out + NODE_OUT_ELEMS;           // (32,390,390,32)
  float* mask_out = edge_out + (size_t)EDGE_OUT_ELEMS;   // (32,390,390)

  float* ab = (float*)d_ws;  // alpha[32] | beta[32]

  fold_kernel<<<1, 32, 0, stream>>>(pw_W, pw_b, edge_W, edge_b, ab);

  // M = 32*390 = 12480 rows -> 780 tiles of 16
  node_wmma_kernel<<<780, 32, 0, stream>>>(b0, b1, b2, b3, pb_W, pb_b,
                                           node_W, node_b, pos_embed, node_out);

  // 32*390*390 edges * 8 lanes / 256 threads = 152100 blocks
  edge_kernel<<<152100, 256, 0, stream>>>(w0, w1, w2, w3, ab, edge_out, mask_out);
}